// ClustGeoEdgeEncoder_55611236548664
// MI455X (gfx1250) — compile-verified
//
#include <hip/hip_runtime.h>
#include <hip/hip_bf16.h>
#include <math.h>

// ---------------------------------------------------------------------------
// ClustGeoEdgeEncoder for MI455X (gfx1250, wave32).
//   Stage 1: gather voxel xyz through clusts into two padded float4 tables
//            (raw xyz0 and -2*xyz0) plus per-point squared norms.
//   Stage 2: one wave per edge. 128x128 distance matrix produced DIRECTLY by
//            64x V_WMMA_F32_16X16X4_F32 tiles: C seeded with n1[i]+n2[j],
//            A = -2*x1 rows, B = x2 cols  =>  D = n1+n2-2*x1.x2 = d^2.
//            Argmin fused via monotonic u64 keys; the >=0 clamp is done as a
//            signed-int max (1x v_max_i32, no NaN canonicalize), the key
//            update is a single v_min_u64. Wave32 shfl_xor butterfly, lane 0
//            emits the 2x19 feature rows.
// ---------------------------------------------------------------------------

typedef float v2f __attribute__((ext_vector_type(2)));
typedef float v8f __attribute__((ext_vector_type(8)));

#define N_VOX    200000
#define N_COLSD  5
#define N_CLUSTS 1024
#define PTSC     128
#define N_EDGES  8192

__global__ __launch_bounds__(256)
void gather_pts_kernel(const float* __restrict__ data,
                       const int*   __restrict__ clusts,
                       float4*      __restrict__ pts,    // [N_CLUSTS*PTSC] xyz0
                       float4*      __restrict__ ptsm2,  // [N_CLUSTS*PTSC] -2*xyz0
                       float*       __restrict__ nrm)    // [N_CLUSTS*PTSC] |p|^2
{
    int t = blockIdx.x * blockDim.x + threadIdx.x;
    if (t >= N_CLUSTS * PTSC) return;
    int v = clusts[t];
    const float* row = data + (size_t)v * N_COLSD;
    float x = row[1], y = row[2], z = row[3];
    pts[t]   = make_float4(x, y, z, 0.0f);
    ptsm2[t] = make_float4(-2.0f * x, -2.0f * y, -2.0f * z, 0.0f);
    nrm[t]   = x * x + y * y + z * z;
}

__global__ __launch_bounds__(32)
void edge_feats_kernel(const int*    __restrict__ edge_index, // [2*N_EDGES]
                       const float4* __restrict__ pts,
                       const float4* __restrict__ ptsm2,
                       const float*  __restrict__ nrm,
                       float*        __restrict__ out)        // [N_EDGES*38]
{
    const int e    = blockIdx.x;
    const int lane = threadIdx.x;        // wave32: 0..31
    const int c1 = edge_index[e];
    const int c2 = edge_index[N_EDGES + e];

    const float* a1 = (const float*)(ptsm2 + (size_t)c1 * PTSC); // A: -2*x1
    const float* p2 = (const float*)(pts   + (size_t)c2 * PTSC); // B:  x2
    const float* n1 = nrm + (size_t)c1 * PTSC;
    const float* n2 = nrm + (size_t)c2 * PTSC;

    const int half = lane >> 4;          // frag half: K = 2*half, 2*half+1
    const int m    = lane & 15;          // row/col within 16-tile

    // Preload all 8 B fragments (x2 column tiles) and their norms.
    v2f   bfrag[8];
    float n2v[8];
#pragma unroll
    for (int tj = 0; tj < 8; ++tj) {
        bfrag[tj] = *(const v2f*)(p2 + (size_t)((tj * 16 + m) * 4 + 2 * half));
        n2v[tj]   = n2[tj * 16 + m];     // N = lane%16 for every D row r
    }

    // Monotonic argmin key: (bits of clamped d2) << 32 | flat index.
    unsigned long long bestkey = 0xFFFFFFFFFFFFFFFFull;

#pragma unroll
    for (int ti = 0; ti < 8; ++ti) {
        // A fragment: rows ti*16 + m of -2*x1, K = 2*half .. 2*half+1
        v2f afrag = *(const v2f*)(a1 + (size_t)((ti * 16 + m) * 4 + 2 * half));
        // n1 for D rows M = 8*half + r, r = 0..7
        float n1v[8];
        const float* n1p = n1 + ti * 16 + 8 * half;
#pragma unroll
        for (int r = 0; r < 8; ++r) n1v[r] = n1p[r];

#pragma unroll
        for (int tj = 0; tj < 8; ++tj) {
            // Seed accumulator with n1[i] + n2[j]; WMMA adds -2*x1.x2.
            v8f acc;
#pragma unroll
            for (int r = 0; r < 8; ++r) acc[r] = n1v[r] + n2v[tj];
            acc = __builtin_amdgcn_wmma_f32_16x16x4_f32(
                false, afrag, false, bfrag[tj], (short)0, acc, false, false);

            // base bits 0-6 and 10-13; r*128 bits 7-9: disjoint -> OR.
            const unsigned base =
                (unsigned)((ti * 16 + 8 * half) * 128 + tj * 16 + m);
#pragma unroll
            for (int r = 0; r < 8; ++r) {
                // d2 >= 0 mathematically; clamp cancellation noise in the
                // integer domain: bits of non-negative f32 are monotonic and
                // non-negative as signed int, so smax(bits,0) == bits of
                // max(d2, +0.0). One v_max_i32, no NaN canonicalize.
                int cb = __float_as_int(acc[r]);
                cb = (cb < 0) ? 0 : cb;
                unsigned long long key =
                    ((unsigned long long)(unsigned)cb << 32)
                    | (unsigned long long)(base | (unsigned)(r * 128));
                bestkey = (key < bestkey) ? key : bestkey;
            }
        }
    }

    // Wave32 butterfly argmin; u64 key compare gives value order with
    // first-occurrence (smallest flat index) tie-break for free.
#pragma unroll
    for (int off = 16; off >= 1; off >>= 1) {
        unsigned hi = (unsigned)(bestkey >> 32);
        unsigned lo = (unsigned)(bestkey & 0xFFFFFFFFull);
        unsigned ohi = (unsigned)__shfl_xor((int)hi, off, 32);
        unsigned olo = (unsigned)__shfl_xor((int)lo, off, 32);
        unsigned long long okey = ((unsigned long long)ohi << 32)
                                  | (unsigned long long)olo;
        bestkey = (okey < bestkey) ? okey : bestkey;
    }

    if (lane == 0) {
        const unsigned bidx = (unsigned)(bestkey & 0xFFFFFFFFull);
        const int i1 = (int)(bidx >> 7);    // / 128
        const int i2 = (int)(bidx & 127u);  // % 128
        float4 v1 = pts[(size_t)c1 * PTSC + i1];
        float4 v2 = pts[(size_t)c2 * PTSC + i2];

        float dx = v1.x - v2.x, dy = v1.y - v2.y, dz = v1.z - v2.z;
        float l2   = dx * dx + dy * dy + dz * dz;
        float lend = sqrtf(l2);
        float inv  = (lend > 0.0f) ? (1.0f / lend) : 1.0f; // lend==0 -> disp
        float nx = dx * inv, ny = dy * inv, nz = dz * inv;

        float B[9] = { nx * nx, nx * ny, nx * nz,
                       ny * nx, ny * ny, ny * nz,
                       nz * nx, nz * ny, nz * nz };

        float* o = out + (size_t)e * 38;
        // feats: [v1, v2, dispn, lend, B]
        o[0] = v1.x; o[1] = v1.y; o[2] = v1.z;
        o[3] = v2.x; o[4] = v2.y; o[5] = v2.z;
        o[6] = nx;   o[7] = ny;   o[8] = nz;
        o[9] = lend;
#pragma unroll
        for (int k = 0; k < 9; ++k) o[10 + k] = B[k];
        // feats_flip: [v2, v1, -dispn, lend, B]
        o[19] = v2.x; o[20] = v2.y; o[21] = v2.z;
        o[22] = v1.x; o[23] = v1.y; o[24] = v1.z;
        o[25] = -nx;  o[26] = -ny;  o[27] = -nz;
        o[28] = lend;
#pragma unroll
        for (int k = 0; k < 9; ++k) o[29 + k] = B[k];
    }
}

extern "C" void kernel_launch(void* const* d_in, const int* in_sizes, int n_in,
                              void* d_out, int out_size, void* d_ws, size_t ws_size,
                              hipStream_t stream) {
    const float* data       = (const float*)d_in[0];
    const int*   clusts     = (const int*)d_in[1];
    const int*   edge_index = (const int*)d_in[2];
    float*       out        = (float*)d_out;

    // Workspace: pts (2 MB) | ptsm2 (2 MB) | norms (0.5 MB)  = 4.5 MB
    const size_t tbl = (size_t)N_CLUSTS * PTSC * sizeof(float4);
    float4* pts   = (float4*)d_ws;
    float4* ptsm2 = (float4*)((char*)d_ws + tbl);
    float*  nrm   = (float*)((char*)d_ws + 2 * tbl);

    gather_pts_kernel<<<(N_CLUSTS * PTSC + 255) / 256, 256, 0, stream>>>(
        data, clusts, pts, ptsm2, nrm);
    edge_feats_kernel<<<N_EDGES, 32, 0, stream>>>(
        edge_index, pts, ptsm2, nrm, out);
}